// BatchHardTriplet_60997125538163
// MI455X (gfx1250) — compile-verified
//
#include <hip/hip_runtime.h>
#include <hip/hip_bf16.h>

typedef __attribute__((ext_vector_type(2))) float v2f;
typedef __attribute__((ext_vector_type(4))) float v4f;
typedef __attribute__((ext_vector_type(8))) float v8f;

#define BATCH 8192
#define DIM 128
#define MARGIN_F 0.3f
#define COL_SPLITS 8
#define COLS_PER_SPLIT (BATCH / COL_SPLITS)     // 1024
#define TILES_PER_SPLIT (COLS_PER_SPLIT / 16)   // 64
#define ROWS_PER_BLOCK 128
#define BIGF 1e30f

// ---------------------------------------------------------------------------
// Kernel 0: hoist the reference's elementwise clamp on the transposed operand
// out of the O(B^2*D) loop: embc = clamp(emb, -1, 1). O(B*D), 4 MB, trivial.
// ---------------------------------------------------------------------------
__global__ __launch_bounds__(256)
void bh_clamp(const float* __restrict__ emb, float* __restrict__ embc) {
  const int i = (blockIdx.x * 256 + threadIdx.x) * 4;   // BATCH*DIM is a multiple of 1024
  v4f v = *(const v4f*)(emb + i);
  v.x = __builtin_amdgcn_fmed3f(v.x, -1.0f, 1.0f);
  v.y = __builtin_amdgcn_fmed3f(v.y, -1.0f, 1.0f);
  v.z = __builtin_amdgcn_fmed3f(v.z, -1.0f, 1.0f);
  v.w = __builtin_amdgcn_fmed3f(v.w, -1.0f, 1.0f);
  *(v4f*)(embc + i) = v;
}

__device__ __forceinline__ void load_chunk16(v2f* __restrict__ b, const float* __restrict__ brow, int koff) {
#pragma unroll
  for (int t = 0; t < 16; ++t)
    b[t] = *(const v2f*)(brow + koff + 4 * t);
}

__device__ __forceinline__ v8f compute_chunk16(const v2f* __restrict__ a, const v2f* __restrict__ b, v8f acc) {
#pragma unroll
  for (int t = 0; t < 16; ++t) {
    acc = __builtin_amdgcn_wmma_f32_16x16x4_f32(
        /*neg_a=*/false, a[t], /*neg_b=*/false, b[t],
        /*c_mod=*/(short)0, acc, /*reuse_a=*/false, /*reuse_b=*/false);
  }
  return acc;
}

// ---------------------------------------------------------------------------
// Kernel 1: one wave owns a 16-row strip of the sim matrix and sweeps one
// column split, fusing the masked row-wise min/max reduction so the 8192x8192
// sim matrix never leaves registers. B panels (pre-clamped) are software-
// pipelined at half-tile (K=64) granularity so ~16 global_load_b64 stay in
// flight under each 16-WMMA chain.
// ---------------------------------------------------------------------------
__global__ __launch_bounds__(256)
void bh_sim_partial(const float* __restrict__ emb, const float* __restrict__ embc,
                    const int* __restrict__ labels,
                    float* __restrict__ posmin_part, float* __restrict__ negmax_part) {
  const int lane   = threadIdx.x & 31;
  const int waveId = threadIdx.x >> 5;
  const int lane15 = lane & 15;
  const int hi     = lane >> 4;                 // 0: K pair {0,1} / C rows 0-7; 1: K pair {2,3} / C rows 8-15
  const int row_base = blockIdx.x * ROWS_PER_BLOCK + waveId * 16;
  const int csplit   = blockIdx.y;
  const int col0     = csplit * COLS_PER_SPLIT;

  // A operand (unclamped emb, per the reference), resident across the sweep.
  // ISA A-layout (16x4 f32): lane&15 = M, VGPR pair holds K = 4t + 2*hi .. +1
  v2f a[32];
  const float* arow = emb + (size_t)(row_base + lane15) * DIM + 2 * hi;
#pragma unroll
  for (int t = 0; t < 32; ++t)
    a[t] = *(const v2f*)(arow + 4 * t);

  // Labels for the 8 rows this lane-half accumulates (C layout: vgpr m -> M = hi*8 + m)
  int li[8];
#pragma unroll
  for (int m = 0; m < 8; ++m)
    li[m] = labels[row_base + hi * 8 + m];

  float posmin[8], negmax[8];
#pragma unroll
  for (int m = 0; m < 8; ++m) { posmin[m] = BIGF; negmax[m] = -BIGF; }

  // Pipeline prologue: B chunk 0 + column label of tile 0
  v2f b0[16], b1[16];
  const float* brow = embc + (size_t)(col0 + lane15) * DIM + 2 * hi;
  load_chunk16(b0, brow, 0);
  int lj = labels[col0 + lane15];

  for (int ct = 0; ct < TILES_PER_SPLIT; ++ct) {
    const int col_base = col0 + ct * 16;
    const int selfm = col_base + lane15 - row_base - hi * 8;  // element m is self iff m == selfm

    // next-tile pointers (clamped in-bounds; results unused on last iteration)
    const int ct_n = (ct + 1 < TILES_PER_SPLIT) ? (ct + 1) : 0;
    const float* brow_n = embc + (size_t)(col0 + ct_n * 16 + lane15) * DIM + 2 * hi;

    v8f acc = {};
    // K = 64..127 loads fly under the K = 0..63 WMMA chain
    load_chunk16(b1, brow, 64);
    const int lj_n = labels[col0 + ct_n * 16 + lane15];
    acc = compute_chunk16(a, b0, acc);
    // next tile's K = 0..63 loads fly under the K = 64..127 WMMA chain
    load_chunk16(b0, brow_n, 0);
    acc = compute_chunk16(a + 16, b1, acc);

    // Masked running reductions (sim-space: pos -> min, neg -> max)
#pragma unroll
    for (int m = 0; m < 8; ++m) {
      const float s   = acc[m];
      const bool same = (li[m] == lj);
      const bool self = (m == selfm);
      const float sp  = (same && !self) ? s : BIGF;
      const float sn  = (!same)         ? s : -BIGF;
      posmin[m] = fminf(posmin[m], sp);
      negmax[m] = fmaxf(negmax[m], sn);
    }

    lj   = lj_n;
    brow = brow_n;
  }

  // Reduce the 16 columns of each row across the 16 lanes of each half-wave
#pragma unroll
  for (int m = 0; m < 8; ++m) {
    float pm = posmin[m], nm = negmax[m];
#pragma unroll
    for (int off = 1; off < 16; off <<= 1) {
      pm = fminf(pm, __shfl_xor(pm, off, 32));
      nm = fmaxf(nm, __shfl_xor(nm, off, 32));
    }
    posmin[m] = pm; negmax[m] = nm;
  }

  if (lane15 == 0) {                             // lanes 0 and 16: rows hi*8 .. hi*8+7
#pragma unroll
    for (int m = 0; m < 8; ++m) {
      const int row = row_base + hi * 8 + m;
      posmin_part[(size_t)csplit * BATCH + row] = posmin[m];
      negmax_part[(size_t)csplit * BATCH + row] = negmax[m];
    }
  }
}

// ---------------------------------------------------------------------------
// Kernel 2: single block combines column splits, per-row loss, mean with the
// reference's cnt>0 fallback.
// ---------------------------------------------------------------------------
__global__ __launch_bounds__(256)
void bh_finalize(const float* __restrict__ posmin_part,
                 const float* __restrict__ negmax_part,
                 float* __restrict__ out) {
  __shared__ float ssum[256];
  __shared__ int   scnt[256];
  const int tid = threadIdx.x;
  float lsum = 0.0f;
  int   lcnt = 0;
  for (int row = tid; row < BATCH; row += 256) {
    float pm = BIGF, nm = -BIGF;
#pragma unroll
    for (int cgrp = 0; cgrp < COL_SPLITS; ++cgrp) {
      pm = fminf(pm, posmin_part[cgrp * BATCH + row]);
      nm = fmaxf(nm, negmax_part[cgrp * BATCH + row]);
    }
    const float hp = fmaxf(1.0f - pm, -1.0f);    // sentinel -> -1 (no positive)
    const float hn = fminf(1.0f - nm, 10.0f);    // sentinel -> 10 (no negative)
    const bool valid = (hp >= 0.0f) && (hn < 10.0f);
    const float per  = fmaxf(0.0f, hp - hn + MARGIN_F);
    lsum += valid ? per : 0.0f;
    lcnt += valid ? 1 : 0;
  }
  ssum[tid] = lsum;
  scnt[tid] = lcnt;
  __syncthreads();
  for (int s = 128; s > 0; s >>= 1) {
    if (tid < s) { ssum[tid] += ssum[tid + s]; scnt[tid] += scnt[tid + s]; }
    __syncthreads();
  }
  if (tid == 0) {
    const float sum = ssum[0];
    const int   cnt = scnt[0];
    out[0] = (cnt > 0) ? (sum / (float)cnt) : 0.0f;
  }
}

extern "C" void kernel_launch(void* const* d_in, const int* in_sizes, int n_in,
                              void* d_out, int out_size, void* d_ws, size_t ws_size,
                              hipStream_t stream) {
  const float* emb    = (const float*)d_in[0];
  const int*   labels = (const int*)d_in[1];
  float* out = (float*)d_out;

  float* embc        = (float*)d_ws;                                   // [BATCH*DIM] clamped copy
  float* posmin_part = embc + (size_t)BATCH * DIM;                     // [COL_SPLITS][BATCH]
  float* negmax_part = posmin_part + (size_t)COL_SPLITS * BATCH;       // [COL_SPLITS][BATCH]

  bh_clamp<<<(BATCH * DIM) / (256 * 4), 256, 0, stream>>>(emb, embc);

  dim3 grid(BATCH / ROWS_PER_BLOCK, COL_SPLITS);
  bh_sim_partial<<<grid, 256, 0, stream>>>(emb, embc, labels, posmin_part, negmax_part);
  bh_finalize<<<1, 256, 0, stream>>>(posmin_part, negmax_part, out);
}